// OscillatorCore_3255585210525
// MI455X (gfx1250) — compile-verified
//
#include <hip/hip_runtime.h>
#include <cstdint>

#define EPSF   1e-6f
#define BETA   0.2f
#define DTF    0.125f
#define NSTEPS 8
#define TOW    4
#define DIM    512
#define EPB    8   // batch elements per block (one per wave32)

typedef unsigned int u32;
typedef u32 v4u __attribute__((ext_vector_type(4)));
typedef int  v8i __attribute__((ext_vector_type(8)));
typedef int  v4i __attribute__((ext_vector_type(4)));

// ---------- small float4 helpers ----------
__device__ __forceinline__ float4 f4_sub(float4 a, float4 b) {
    return make_float4(a.x - b.x, a.y - b.y, a.z - b.z, a.w - b.w);
}
__device__ __forceinline__ float f4_dot(float4 a, float4 b) {
    return fmaf(a.x, b.x, fmaf(a.y, b.y, fmaf(a.z, b.z, a.w * b.w)));
}
__device__ __forceinline__ float4 f4_fma(float4 a, float s, float4 c) {
    return make_float4(fmaf(a.x, s, c.x), fmaf(a.y, s, c.y),
                       fmaf(a.z, s, c.z), fmaf(a.w, s, c.w));
}

// wave32 XOR butterfly all-reduce: 1 ds_swizzle + 1 v_add per stage
__device__ __forceinline__ float sw_add(float v, int imm_pattern) {
    // caller passes compile-time constant patterns only
    return v;
}
__device__ __forceinline__ float wave_sum(float v) {
    v += __int_as_float(__builtin_amdgcn_ds_swizzle(__float_as_int(v), 0x401F)); // xor 16
    v += __int_as_float(__builtin_amdgcn_ds_swizzle(__float_as_int(v), 0x201F)); // xor 8
    v += __int_as_float(__builtin_amdgcn_ds_swizzle(__float_as_int(v), 0x101F)); // xor 4
    v += __int_as_float(__builtin_amdgcn_ds_swizzle(__float_as_int(v), 0x081F)); // xor 2
    v += __int_as_float(__builtin_amdgcn_ds_swizzle(__float_as_int(v), 0x041F)); // xor 1
    return v;
}

// ---------- TDM: 1-D contiguous tile (nelem f32) global -> LDS ----------
__device__ __forceinline__ void tdm_load_1d(const void* gptr, u32 lds_off_bytes, u32 nelem) {
    unsigned long long ga = (unsigned long long)(uintptr_t)gptr;
    // D# group 0 (128b): count=1 | lds_addr | global_addr[56:0] | type=2
    v4u g0;
    g0[0] = 1u;                                            // count=1 (valid), user mode
    g0[1] = lds_off_bytes;                                 // lds_addr (bytes)
    g0[2] = (u32)(ga & 0xFFFFFFFFull);                     // global_addr[31:0]
    g0[3] = (u32)((ga >> 32) & 0x1FFFFFFull) | (2u << 30); // global_addr[56:32], type=2
    // D# group 1 (256b): data_size=4B, tensor_dim0 = tile_dim0 = nelem, stride0 = nelem
    v8i g1;
    g1[0] = (int)(2u << 16);                               // workgroup_mask=0, data_size=2 (4B)
    g1[1] = (int)((nelem & 0xFFFFu) << 16);                // tensor_dim0[15:0]
    g1[2] = (int)(((nelem >> 16) & 0xFFFFu) | (1u << 16)); // tensor_dim0[31:16], tensor_dim1=1
    g1[3] = (int)((nelem & 0xFFFFu) << 16);                // tensor_dim1[31:16]=0, tile_dim0
    g1[4] = 0;                                             // tile_dim1=0, tile_dim2=0 (1-D tile)
    g1[5] = (int)nelem;                                    // tensor_dim0_stride[31:0]
    g1[6] = 0;                                             // stride0[47:32]=0, stride1[15:0]=0
    g1[7] = 0;
    v4i z2 = {0, 0, 0, 0};
    v4i z3 = {0, 0, 0, 0};
#if __clang_major__ >= 23
    v8i z8 = {0, 0, 0, 0, 0, 0, 0, 0};
    __builtin_amdgcn_tensor_load_to_lds(g0, g1, z2, z3, z8, 0);
#else
    __builtin_amdgcn_tensor_load_to_lds(g0, g1, z2, z3, 0);
#endif
}

extern __shared__ float smem[]; // [EPB * TOW * DIM] floats = 64 KB (tower_neg slices)

__global__ __launch_bounds__(256, 1)
void OscillatorCore_kernel(const float* __restrict__ x_init,
                           const float* __restrict__ x_ref,
                           const float* __restrict__ tower_pos,
                           const float* __restrict__ tower_neg,
                           float* __restrict__ out) {
    const int wave   = (int)(threadIdx.x >> 5);
    const int lane   = (int)(threadIdx.x & 31);
    const int wave_s = __builtin_amdgcn_readfirstlane(wave); // scalar for D# fields
    const int b0     = (int)blockIdx.x * EPB;
    const int b      = b0 + wave;

    // --- TDM engine: stream this element's tower_neg slice into private LDS ---
    const size_t gofs = ((size_t)b0 + (size_t)wave_s) * (TOW * DIM);
    tdm_load_1d(tower_neg + gofs, (u32)(wave_s * TOW * DIM * sizeof(float)), TOW * DIM);

    // --- VMEM engine (overlapping TDM): x0, x_ref, and all tower_pos chunks ---
    // lane layout: d = 128*i + 4*lane + j  (coalesced b128 per chunk i)
    float4 xr[4], u0[4], pos[TOW][4];
    const size_t base = (size_t)b * DIM + (size_t)(lane * 4);
    float nu = 0.f;
#pragma unroll
    for (int i = 0; i < 4; ++i) {
        float4 x0 = *reinterpret_cast<const float4*>(x_init + base + i * 128);
        xr[i]     = *reinterpret_cast<const float4*>(x_ref  + base + i * 128);
        u0[i]     = f4_sub(xr[i], x0);
        nu       += f4_dot(u0[i], u0[i]);
    }
    const float* pbase = tower_pos + gofs + (size_t)(lane * 4);
#pragma unroll
    for (int t = 0; t < TOW; ++t)
#pragma unroll
        for (int i = 0; i < 4; ++i)
            pos[t][i] = *reinterpret_cast<const float4*>(pbase + t * DIM + i * 128);

    nu = wave_sum(nu);
    const float N0 = sqrtf(nu); // ||u0||  (exact: ||u_n|| = |alpha_n| * N0)

    // --- consume towers: D_t = (delta_t . u0) / (||delta_t|| + eps) ---
    __builtin_amdgcn_s_wait_tensorcnt(0);
    float Dt[TOW];
#pragma unroll
    for (int t = 0; t < TOW; ++t) {
        const int off = wave_s * (TOW * DIM) + t * DIM + lane * 4;
        float dd = 0.f, du = 0.f;
#pragma unroll
        for (int i = 0; i < 4; ++i) {
            float4 n = *reinterpret_cast<const float4*>(&smem[off + i * 128]);
            float4 d = f4_sub(pos[t][i], n);
            dd += f4_dot(d, d);
            du += f4_dot(d, u0[i]);
        }
        dd = wave_sum(dd);
        du = wave_sum(du);
        Dt[t] = du * __builtin_amdgcn_rcpf(sqrtf(dd) + EPSF);
    }

    // --- exact scalar collapse of the 8-step scan (v0=0 => state in span{u0}) ---
    // u_n = alpha_n*u0, v_n = gamma_n*u0 ; q = |alpha|*N0 + eps
    // cos_t = Dt*alpha/q ; S = sum cos*sigmoid(6 cos)
    // A = -beta*gamma + alpha + S*alpha/q ; gamma += DT*A ; alpha -= DT*gamma
    float alpha = 1.f, gamma = 0.f;
#pragma unroll
    for (int s = 0; s < NSTEPS; ++s) {
        const float q     = fmaf(fabsf(alpha), N0, EPSF);
        const float ainvq = alpha * __builtin_amdgcn_rcpf(q);
        float S = 0.f;
#pragma unroll
        for (int t = 0; t < TOW; ++t) {
            const float c = Dt[t] * ainvq;
            S += c * __builtin_amdgcn_rcpf(1.f + __expf(-6.f * c));
        }
        const float A = fmaf(-BETA, gamma, alpha) + S * ainvq;
        gamma = fmaf(DTF, A, gamma);
        alpha = fmaf(-DTF, gamma, alpha);
    }

    // --- x_final = x_ref - alpha_final * u0 ---
#pragma unroll
    for (int i = 0; i < 4; ++i)
        *reinterpret_cast<float4*>(out + base + i * 128) = f4_fma(u0[i], -alpha, xr[i]);
}

extern "C" void kernel_launch(void* const* d_in, const int* in_sizes, int n_in,
                              void* d_out, int out_size, void* d_ws, size_t ws_size,
                              hipStream_t stream) {
    const float* x_init    = (const float*)d_in[0];
    const float* x_ref     = (const float*)d_in[1];
    const float* tower_pos = (const float*)d_in[2];
    const float* tower_neg = (const float*)d_in[3];
    float*       out       = (float*)d_out;

    const int B = in_sizes[0] / DIM; // 32768
    const size_t shmem = (size_t)EPB * TOW * DIM * sizeof(float); // 64 KB

    (void)hipFuncSetAttribute((const void*)OscillatorCore_kernel,
                              hipFuncAttributeMaxDynamicSharedMemorySize,
                              (int)shmem);

    dim3 grid((unsigned)(B / EPB));
    dim3 block(256);
    hipLaunchKernelGGL(OscillatorCore_kernel, grid, block, shmem, stream,
                       x_init, x_ref, tower_pos, tower_neg, out);
}